// EntmaxBisectLoss_66451734003726
// MI455X (gfx1250) — compile-verified
//
#include <hip/hip_runtime.h>
#include <stdint.h>

// ---------------------------------------------------------------------------
// EntmaxBisectLoss for MI455X (gfx1250, wave32)
//   X: [4096, 32000] fp32, target: [4096] i32 -> scalar fp32 mean loss
//
// One row per 1024-thread block iteration; row held in registers
// (32 regs/thread) for the entire 50-step bisection; one LDS barrier per
// bisection step; next row prefetched into LDS by the CDNA5 Tensor Data
// Mover (tensor_load_to_lds, TENSORcnt) overlapping the current row's
// compute.
// ---------------------------------------------------------------------------

#define N_SAMPLES 4096
#define N_CLASSES 32000
#define THREADS   1024
#define WAVES     32
#define EPT       32                    // 32 * 1024 = 32768 slots >= 32000
#define GRID      1024                  // 4 rows per block (grid-stride)

#define NEG_HUGE (-3.402823466e38f)     // finite pad: relu(.-tau)=0, no NaN

typedef unsigned int u32x4 __attribute__((ext_vector_type(4)));
typedef unsigned int u32x8 __attribute__((ext_vector_type(8)));

// ---- wave / block reductions (wave32) -------------------------------------

__device__ __forceinline__ float waveSum(float v) {
#pragma unroll
  for (int off = 16; off; off >>= 1) v += __shfl_xor(v, off, 32);
  return v;
}

__device__ __forceinline__ float waveMax(float v) {
#pragma unroll
  for (int off = 16; off; off >>= 1) v = fmaxf(v, __shfl_xor(v, off, 32));
  return v;
}

// One barrier. Every wave re-reduces the 32 partials itself, so the total is
// available in all lanes with no second barrier / broadcast.
__device__ __forceinline__ float blockSum(float v, float* buf, int wave, int lane) {
  v = waveSum(v);
  if (lane == 0) buf[wave] = v;
  __syncthreads();
  return waveSum(buf[lane]);
}

__device__ __forceinline__ float blockMax(float v, float* buf, int wave, int lane) {
  v = waveMax(v);
  if (lane == 0) buf[wave] = v;
  __syncthreads();
  return waveMax(buf[lane]);
}

// ---- CDNA5 Tensor Data Mover: one descriptor stages a whole row -----------
// D# layout per CDNA5 ISA:
//   group0: count=1 | lds_addr | global_addr[56:0] | type=2
//   group1: data_size=4B, tensor_dim0 = tile_dim0 = 32000, tile_dim1 = 1,
//           tensor_dim1 = 1, tensor_dim0_stride = 32000  (2D tile, 128 KB)
// VADDR2/VADDR3 omitted (NULL) -> 2-group form, tensors up to 2D.
// Low 32 bits of the flat shared pointer are the LDS byte offset
// (ISA 10.2 aperture mapping).
__device__ __forceinline__ void tdmLoadRow(const float* __restrict__ src,
                                           float* stage) {
  const unsigned long long ga = (unsigned long long)(uintptr_t)src;
  u32x4 g0;
  g0[0] = 1u;                                    // count=1, user descriptor
  g0[1] = (unsigned)(uintptr_t)stage;            // lds_addr
  g0[2] = (unsigned)ga;                          // global_addr[31:0]
  g0[3] = (unsigned)(ga >> 32) | 0x80000000u;    // global_addr[56:32]|type=2
  u32x8 g1;
  g1[0] = 0x00020000u;                           // data_size=2 (4 bytes)
  g1[1] = ((unsigned)N_CLASSES & 0xFFFFu) << 16; // tensor_dim0[15:0]
  g1[2] = ((unsigned)N_CLASSES >> 16) | (1u << 16); // dim0[31:16]|tensor_dim1=1
  g1[3] = ((unsigned)N_CLASSES & 0xFFFFu) << 16; // tile_dim0 = 32000
  g1[4] = 1u;                                    // tile_dim1=1, tile_dim2=0
  g1[5] = (unsigned)N_CLASSES;                   // tensor_dim0_stride[31:0]
  g1[6] = 0u;
  g1[7] = 0u;
  asm volatile("tensor_load_to_lds %0, %1" :: "s"(g0), "s"(g1) : "memory");
}

__device__ __forceinline__ void tdmWait() {
#if __has_builtin(__builtin_amdgcn_s_wait_tensorcnt)
  __builtin_amdgcn_s_wait_tensorcnt(0);
#else
  asm volatile("s_wait_tensorcnt 0x0" ::: "memory");
#endif
}

// ---- main per-row kernel ---------------------------------------------------

__global__ __launch_bounds__(THREADS, 1)
void entmax_bisect_rows(const float* __restrict__ X,
                        const int*   __restrict__ tgt,
                        float*       __restrict__ rowLoss) {
  __shared__ float stage[EPT * THREADS];      // 128 KB row staging buffer
  __shared__ float rMax[WAVES], rFlo[WAVES];
  __shared__ float rIt[2][WAVES];             // double-buffered: 1 barrier/iter
  __shared__ float rS2[WAVES], rS3[WAVES], rSX[WAVES];

  const int tid  = threadIdx.x;
  const int wave = tid >> 5;
  const int lane = tid & 31;

  // dm0 = tau_hi - tau_lo = 1 - (1/d)^(alpha-1) = 1 - 1/sqrt(32000)
  const float DM0 = 1.0f - 0.00559016994f;

  // prefetch the first row this block owns (wave 0 issues the DMA)
  if (wave == 0)
    tdmLoadRow(X + (size_t)blockIdx.x * N_CLASSES, stage);

  for (int row = blockIdx.x; row < N_SAMPLES; row += GRID) {
    if (wave == 0) tdmWait();   // DMA done (TENSORcnt tracked by wave 0)
    __syncthreads();            // publish LDS to all waves

    // LDS -> registers, Xa = 0.5 * X, pad with -FLT_MAX (stride-1024: no
    // bank conflicts; pads contribute 0 to every relu^2 sum)
    float xa[EPT];
#pragma unroll
    for (int e = 0; e < EPT; ++e) {
      int idx = (e << 10) + tid;
      xa[e] = (idx < N_CLASSES) ? 0.5f * stage[idx] : NEG_HUGE;
    }
    __syncthreads();            // everyone done reading stage

    // overlap: TDM streams the next row into LDS while we bisect this one
    if (wave == 0 && row + GRID < N_SAMPLES)
      tdmLoadRow(X + (size_t)(row + GRID) * N_CLASSES, stage);

    // ---- row max of Xa ----
    float m = NEG_HUGE;
#pragma unroll
    for (int e = 0; e < EPT; ++e) m = fmaxf(m, xa[e]);
    m = blockMax(m, rMax, wave, lane);

    float tau_lo = m - 1.0f;

    // ---- f_lo = sum relu(Xa - tau_lo)^2 - 1 ----
    {
      float a0 = 0.f, a1 = 0.f, a2 = 0.f, a3 = 0.f;
#pragma unroll
      for (int e = 0; e < EPT; e += 4) {
        float t0 = fmaxf(xa[e + 0] - tau_lo, 0.f);
        float t1 = fmaxf(xa[e + 1] - tau_lo, 0.f);
        float t2 = fmaxf(xa[e + 2] - tau_lo, 0.f);
        float t3 = fmaxf(xa[e + 3] - tau_lo, 0.f);
        a0 = fmaf(t0, t0, a0); a1 = fmaf(t1, t1, a1);
        a2 = fmaf(t2, t2, a2); a3 = fmaf(t3, t3, a3);
      }
      float f_lo = blockSum((a0 + a1) + (a2 + a3), rFlo, wave, lane) - 1.0f;

      // ---- 50 bisection steps; one barrier each ----
      float dm = DM0, tau_m = tau_lo;
      for (int it = 0; it < 50; ++it) {
        dm *= 0.5f;
        tau_m = tau_lo + dm;
        float b0 = 0.f, b1 = 0.f, b2 = 0.f, b3 = 0.f;
#pragma unroll
        for (int e = 0; e < EPT; e += 4) {
          float t0 = fmaxf(xa[e + 0] - tau_m, 0.f);
          float t1 = fmaxf(xa[e + 1] - tau_m, 0.f);
          float t2 = fmaxf(xa[e + 2] - tau_m, 0.f);
          float t3 = fmaxf(xa[e + 3] - tau_m, 0.f);
          b0 = fmaf(t0, t0, b0); b1 = fmaf(t1, t1, b1);
          b2 = fmaf(t2, t2, b2); b3 = fmaf(t3, t3, b3);
        }
        float f_m = blockSum((b0 + b1) + (b2 + b3), rIt[it & 1], wave, lane) - 1.0f;
        if (f_m * f_lo >= 0.0f) tau_lo = tau_m;   // uniform across block
      }

      // ---- final pass at last midpoint tau_m:
      //   s2 = sum t^2, s3 = sum t^3, sx = sum t^2 * Xa   (t = relu(Xa-tau_m))
      float s2 = 0.f, s3 = 0.f, sx = 0.f;
#pragma unroll
      for (int e = 0; e < EPT; ++e) {
        float t  = fmaxf(xa[e] - tau_m, 0.f);
        float t2 = t * t;
        s2 += t2;
        s3 = fmaf(t2, t, s3);
        sx = fmaf(t2, xa[e], sx);
      }
      s2 = blockSum(s2, rS2, wave, lane);
      s3 = blockSum(s3, rS3, wave, lane);
      sx = blockSum(sx, rSX, wave, lane);

      if (tid == 0) {
        // p_j = t_j^2 / s2 ; sum p^1.5 = s3 / s2^1.5 ; X = 2*Xa
        float Xt     = X[(size_t)row * N_CLASSES + tgt[row]];
        float inv    = 1.0f / s2;
        float sum_pa = s3 * inv / sqrtf(s2);
        float omega  = (1.0f - sum_pa) * (1.0f / 0.75f);   // alpha*(alpha-1)
        rowLoss[row] = omega + 2.0f * sx * inv - Xt;
      }
    }
  }
}

// ---- deterministic final mean over 4096 row losses ------------------------

__global__ __launch_bounds__(THREADS, 1)
void entmax_final_reduce(const float* __restrict__ rowLoss,
                         float*       __restrict__ out) {
  __shared__ float buf[WAVES];
  const int tid = threadIdx.x;
  float a = 0.f;
  for (int i = tid; i < N_SAMPLES; i += THREADS) a += rowLoss[i];
  a = waveSum(a);
  if ((tid & 31) == 0) buf[tid >> 5] = a;
  __syncthreads();
  a = waveSum(buf[tid & 31]);
  if (tid == 0) out[0] = a * (1.0f / (float)N_SAMPLES);
}

// ---------------------------------------------------------------------------

extern "C" void kernel_launch(void* const* d_in, const int* in_sizes, int n_in,
                              void* d_out, int out_size, void* d_ws, size_t ws_size,
                              hipStream_t stream) {
  const float* X   = (const float*)d_in[0];   // [4096, 32000] fp32
  const int*   tgt = (const int*)d_in[1];     // [4096] i32
  float*       out = (float*)d_out;           // scalar fp32
  float*       ws  = (float*)d_ws;            // >= 4096 floats of scratch

  entmax_bisect_rows<<<GRID, THREADS, 0, stream>>>(X, tgt, ws);
  entmax_final_reduce<<<1, THREADS, 0, stream>>>(ws, out);
}